// HybridBERT4RecGNN_18210661335598
// MI455X (gfx1250) — compile-verified
//
#include <hip/hip_runtime.h>
#include <hip/hip_bf16.h>
#include <math.h>

typedef __attribute__((ext_vector_type(2))) float v2f;
typedef __attribute__((ext_vector_type(8))) float v8f;

#define SEQ_LEN 50
#define DMODEL  64
#define NHEAD   2
#define DHEAD   32

__device__ __forceinline__ float gelu_tanh(float x) {
  const float c = 0.7978845608028654f;      // sqrt(2/pi)
  float x3 = x * x * x;
  return 0.5f * x * (1.0f + tanhf(c * (x + 0.044715f * x3)));
}

// ----------------------------- elementwise helpers -----------------------------
__global__ void zero_kernel(float* __restrict__ p, long n) {
  long i = (long)blockIdx.x * blockDim.x + threadIdx.x;
  if (i < n) p[i] = 0.0f;
}

__global__ void copy2_kernel(const float* __restrict__ s, float* __restrict__ a,
                             float* __restrict__ b, long n) {
  long i = (long)blockIdx.x * blockDim.x + threadIdx.x;
  if (i < n) { float v = s[i]; a[i] = v; b[i] = v; }
}

__global__ void acc_add_kernel(float* __restrict__ acc, const float* __restrict__ x, long n) {
  long i = (long)blockIdx.x * blockDim.x + threadIdx.x;
  if (i < n) acc[i] += x[i];
}

// ----------------------------- GNN weighted scatter-add -----------------------------
// out[dst[e], :] += x[src[e], :] * ew[e]; 16 lanes per edge, float4 per lane.
// Node tables are ~25.6MB -> L2 resident; native f32 L2 atomics via relaxed/agent scope.
__global__ void gnn_scatter_kernel(const float* __restrict__ x, const int* __restrict__ src,
                                   const int* __restrict__ dst, const float* __restrict__ ew,
                                   float* __restrict__ out, int E) {
  long t = (long)blockIdx.x * blockDim.x + threadIdx.x;
  int e = (int)(t >> 4);
  if (e >= E) return;
  int d4 = ((int)t & 15) * 4;
  int s = src[e];
  int d = dst[e];
  float w = ew[e];
  const float* xs = x + (long)s * DMODEL + d4;
  float* o = out + (long)d * DMODEL + d4;
#pragma unroll
  for (int k = 0; k < 4; ++k) {
    __hip_atomic_fetch_add(o + k, xs[k] * w, __ATOMIC_RELAXED, __HIP_MEMORY_SCOPE_AGENT);
  }
}

// ----------------------------- fp32 WMMA GEMM -----------------------------
// C[M,Ncols] = epilogue(a_scale*A[M,K] @ B[K,Ncols] + bias)
// Each wave computes 16 rows x 64 cols: one shared A fragment feeds 4 WMMAs per
// k-step (V_WMMA_F32_16X16X4_F32), amortizing A loads 4x and densifying the
// matrix pipe. Fragment layout (CDNA5 ISA 7.12.2, 32-bit):
//   A 16x4: lanes 0-15 -> M=lane, v0=K(k0+0), v1=K(k0+1); lanes 16-31 -> K(k0+2/3)
//   B 4x16: col = lane&15, same K-per-half/VGPR striping as A
//   C/D 16x16: c[j] = C[8*(lane>=16)+j][lane&15]
__global__ void gemm_wmma_f32_kernel(const float* __restrict__ A, const float* __restrict__ B,
                                     const float* __restrict__ bias, float* __restrict__ C,
                                     int M, int K, int Ncols, float a_scale, int fuse_gelu) {
  const int lane = threadIdx.x & 31;
  const int wave = threadIdx.x >> 5;
  const int half = lane >> 4;
  const int l16  = lane & 15;
  const int tileM = blockIdx.y * (blockDim.x >> 5) + wave;  // 16 rows per wave
  const int nBase = blockIdx.x * 64;                        // 4 N-tiles per wave

  int row  = tileM * 16 + l16;
  int rowc = row < M ? row : (M - 1);     // clamp loads; D rows are independent
  const float* Arow = A + (long)rowc * K;
  const float* Bp   = B + (long)nBase + l16;

  v8f c0, c1, c2, c3;
#pragma unroll
  for (int j = 0; j < 8; ++j) { c0[j] = 0.0f; c1[j] = 0.0f; c2[j] = 0.0f; c3[j] = 0.0f; }

  for (int k0 = 0; k0 < K; k0 += 4) {
    int ka = k0 + 2 * half;
    v2f a;
    a.x = Arow[ka] * a_scale;
    a.y = Arow[ka + 1] * a_scale;
    const float* B0 = Bp + (long)ka * Ncols;
    const float* B1 = Bp + (long)(ka + 1) * Ncols;
    v2f b;
    b.x = B0[0];  b.y = B1[0];
    c0 = __builtin_amdgcn_wmma_f32_16x16x4_f32(false, a, false, b, (short)0, c0, false, false);
    b.x = B0[16]; b.y = B1[16];
    c1 = __builtin_amdgcn_wmma_f32_16x16x4_f32(false, a, false, b, (short)0, c1, false, false);
    b.x = B0[32]; b.y = B1[32];
    c2 = __builtin_amdgcn_wmma_f32_16x16x4_f32(false, a, false, b, (short)0, c2, false, false);
    b.x = B0[48]; b.y = B1[48];
    c3 = __builtin_amdgcn_wmma_f32_16x16x4_f32(false, a, false, b, (short)0, c3, false, false);
  }

  const int colBase = nBase + l16;
#define WMMA_EPILOGUE(CC, TT)                                              \
  {                                                                        \
    int col = colBase + (TT) * 16;                                         \
    float bv = bias ? bias[col] : 0.0f;                                    \
    _Pragma("unroll") for (int j = 0; j < 8; ++j) {                        \
      int r = tileM * 16 + 8 * half + j;                                   \
      if (r < M) {                                                         \
        float v = CC[j] + bv;                                              \
        if (fuse_gelu) v = gelu_tanh(v);                                   \
        C[(long)r * Ncols + col] = v;                                      \
      }                                                                    \
    }                                                                      \
  }
  WMMA_EPILOGUE(c0, 0)
  WMMA_EPILOGUE(c1, 1)
  WMMA_EPILOGUE(c2, 2)
  WMMA_EPILOGUE(c3, 3)
#undef WMMA_EPILOGUE
}

// ----------------------------- fusion: seq_emb + pos -----------------------------
__global__ void fuse_kernel(const int* __restrict__ seq, const int* __restrict__ lengths,
                            const float* __restrict__ bert, const float* __restrict__ gout,
                            const float* __restrict__ pos, float* __restrict__ X,
                            int B, int S, long total) {
  long i = (long)blockIdx.x * blockDim.x + threadIdx.x;
  if (i >= total) return;
  int d  = (int)(i & (DMODEL - 1));
  long bs = i >> 6;
  int s = (int)(bs % S);
  int b = (int)(bs / S);
  int len = lengths[b];
  float alpha = (len <= 10) ? 0.3f : ((len >= 50) ? 0.7f : 0.5f);
  int tok = seq[bs];
  X[i] = alpha * bert[(long)tok * DMODEL + d]
       + (1.0f - alpha) * gout[(long)tok * DMODEL + d]
       + pos[s * DMODEL + d];
}

// ----------------------------- attention (per batch*head block) -----------------------------
__global__ void attn_kernel(const float* __restrict__ Q, const float* __restrict__ K,
                            const float* __restrict__ V, const int* __restrict__ seq,
                            float* __restrict__ CTX) {
  int bh = blockIdx.x;
  int b = bh >> 1;        // NHEAD == 2
  int h = bh & 1;
  __shared__ float ks[SEQ_LEN * DHEAD];
  __shared__ float vs[SEQ_LEN * DHEAD];
  __shared__ int   mk[SEQ_LEN];
  const int t = threadIdx.x;  // blockDim.x == 64
  const long base = (long)b * SEQ_LEN * DMODEL + h * DHEAD;

  for (int i = t; i < SEQ_LEN * DHEAD; i += 64) {
    int r = i >> 5, d = i & (DHEAD - 1);
    ks[i] = K[base + (long)r * DMODEL + d];
    vs[i] = V[base + (long)r * DMODEL + d];
  }
  for (int i = t; i < SEQ_LEN; i += 64) mk[i] = (seq[b * SEQ_LEN + i] != 0);
  __syncthreads();

  if (t < SEQ_LEN) {
    const float scale = 0.17677669529663687f;   // 1/sqrt(32)
    float qv[DHEAD];
#pragma unroll
    for (int d = 0; d < DHEAD; ++d) qv[d] = Q[base + (long)t * DMODEL + d];

    float sc[SEQ_LEN];
    float mx = -3.0e38f;
    for (int k = 0; k < SEQ_LEN; ++k) {
      float s = 0.0f;
#pragma unroll
      for (int d = 0; d < DHEAD; ++d) s += qv[d] * ks[k * DHEAD + d];
      s *= scale;
      if (!mk[k]) s = -1.0e9f;
      sc[k] = s;
      mx = fmaxf(mx, s);
    }
    float sum = 0.0f;
    for (int k = 0; k < SEQ_LEN; ++k) { sc[k] = expf(sc[k] - mx); sum += sc[k]; }
    float inv = 1.0f / sum;
    for (int d = 0; d < DHEAD; ++d) {
      float acc = 0.0f;
      for (int k = 0; k < SEQ_LEN; ++k) acc += sc[k] * vs[k * DHEAD + d];
      CTX[base + (long)t * DMODEL + d] = acc * inv;
    }
  }
}

// ----------------------------- residual + layernorm (in-place safe per row) -----------------------------
__global__ void add_ln_kernel(const float* X, const float* Y,
                              const float* __restrict__ g, const float* __restrict__ bt,
                              float* Out, int rows) {
  int r = blockIdx.x * blockDim.x + threadIdx.x;
  if (r >= rows) return;
  float v[DMODEL];
  float m = 0.0f;
#pragma unroll
  for (int d = 0; d < DMODEL; ++d) {
    float t = X[(long)r * DMODEL + d] + Y[(long)r * DMODEL + d];
    v[d] = t; m += t;
  }
  m *= (1.0f / DMODEL);
  float var = 0.0f;
#pragma unroll
  for (int d = 0; d < DMODEL; ++d) { float t = v[d] - m; var += t * t; }
  var *= (1.0f / DMODEL);
  float inv = rsqrtf(var + 1e-5f);
#pragma unroll
  for (int d = 0; d < DMODEL; ++d)
    Out[(long)r * DMODEL + d] = (v[d] - m) * inv * g[d] + bt[d];
}

// ----------------------------- final LN + gather -----------------------------
__global__ void final_kernel(const float* __restrict__ X, const int* __restrict__ lengths,
                             const float* __restrict__ g, const float* __restrict__ bt,
                             float* __restrict__ out, int B) {
  int b = blockIdx.x * blockDim.x + threadIdx.x;
  if (b >= B) return;
  long r = (long)b * SEQ_LEN + (lengths[b] - 1);
  float v[DMODEL];
  float m = 0.0f;
#pragma unroll
  for (int d = 0; d < DMODEL; ++d) { float t = X[r * DMODEL + d]; v[d] = t; m += t; }
  m *= (1.0f / DMODEL);
  float var = 0.0f;
#pragma unroll
  for (int d = 0; d < DMODEL; ++d) { float t = v[d] - m; var += t * t; }
  var *= (1.0f / DMODEL);
  float inv = rsqrtf(var + 1e-5f);
#pragma unroll
  for (int d = 0; d < DMODEL; ++d)
    out[(long)b * DMODEL + d] = (v[d] - m) * inv * g[d] + bt[d];
}

// ----------------------------- host orchestration -----------------------------
extern "C" void kernel_launch(void* const* d_in, const int* in_sizes, int n_in,
                              void* d_out, int out_size, void* d_ws, size_t ws_size,
                              hipStream_t stream) {
  const int*   seq     = (const int*)  d_in[0];
  const int*   lengths = (const int*)  d_in[1];
  const int*   eidx    = (const int*)  d_in[2];
  const float* ew      = (const float*)d_in[3];
  const float* bert    = (const float*)d_in[4];
  const float* gemb    = (const float*)d_in[5];
  const float* pw      = (const float*)d_in[6];
  const float* pb      = (const float*)d_in[7];
  const float* pos     = (const float*)d_in[8];
  const float* wq      = (const float*)d_in[9];
  const float* wk      = (const float*)d_in[10];
  const float* wv      = (const float*)d_in[11];
  const float* wo      = (const float*)d_in[12];
  const float* w1      = (const float*)d_in[13];
  const float* b1      = (const float*)d_in[14];
  const float* w2      = (const float*)d_in[15];
  const float* b2      = (const float*)d_in[16];
  const float* ln1g    = (const float*)d_in[17];
  const float* ln1b    = (const float*)d_in[18];
  const float* ln2g    = (const float*)d_in[19];
  const float* ln2b    = (const float*)d_in[20];
  const float* lnfg    = (const float*)d_in[21];
  const float* lnfb    = (const float*)d_in[22];

  const int  E   = in_sizes[3];
  const int  Nn  = in_sizes[4] / DMODEL;
  const int  Bb  = in_sizes[1];
  const int  Ss  = in_sizes[0] / Bb;   // 50
  const long N64 = (long)Nn * DMODEL;
  const int  T   = Bb * Ss;
  const long T64 = (long)T * DMODEL;

  const int* src = eidx;
  const int* dst = eidx + E;

  float* acc = (float*)d_ws;
  float* xc  = acc + N64;
  float* xn  = xc + N64;
  float* X   = xn + N64;
  float* Qb  = X + T64;
  float* Kb  = Qb + T64;
  float* Vb  = Kb + T64;
  float* CTX = Vb + T64;
  float* HFF = CTX + T64;            // T x 256

  dim3 b256(256);
  auto gsz = [](long n, int bs) { return dim3((unsigned)((n + bs - 1) / bs)); };

  // ---- LightGCN: acc = e0 + e1 + e2 ----
  copy2_kernel<<<gsz(N64, 256), b256, 0, stream>>>(gemb, xc, acc, N64);

  zero_kernel<<<gsz(N64, 256), b256, 0, stream>>>(xn, N64);
  gnn_scatter_kernel<<<gsz((long)E * 16, 256), b256, 0, stream>>>(xc, src, dst, ew, xn, E);
  acc_add_kernel<<<gsz(N64, 256), b256, 0, stream>>>(acc, xn, N64);

  zero_kernel<<<gsz(N64, 256), b256, 0, stream>>>(xc, N64);
  gnn_scatter_kernel<<<gsz((long)E * 16, 256), b256, 0, stream>>>(xn, src, dst, ew, xc, E);
  acc_add_kernel<<<gsz(N64, 256), b256, 0, stream>>>(acc, xc, N64);

  // ---- GEMM launch geometry: 128 threads = 4 waves = 4 M-tiles; blockIdx.x = 64-col group ----
  dim3 gemmBlk(128);
  const int tilesM_node = (Nn + 15) / 16;
  const int tilesM_tok  = (T + 15) / 16;
  dim3 grdNode(1, (tilesM_node + 3) / 4);   // Ncols = 64
  dim3 grdTok64(1, (tilesM_tok + 3) / 4);   // Ncols = 64
  dim3 grdTok256(4, (tilesM_tok + 3) / 4);  // Ncols = 256

  // ---- gnn_out = (acc/3) @ pw + pb  (reuse xn) ----
  float* gout = xn;
  gemm_wmma_f32_kernel<<<grdNode, gemmBlk, 0, stream>>>(
      acc, pw, pb, gout, Nn, 64, 64, 1.0f / 3.0f, 0);

  // ---- fusion + positional ----
  fuse_kernel<<<gsz(T64, 256), b256, 0, stream>>>(seq, lengths, bert, gout, pos, X, Bb, Ss, T64);

  // ---- transformer blocks ----
  for (int i = 0; i < 2; ++i) {
    const float* wqi = wq + (long)i * 64 * 64;
    const float* wki = wk + (long)i * 64 * 64;
    const float* wvi = wv + (long)i * 64 * 64;
    const float* woi = wo + (long)i * 64 * 64;

    gemm_wmma_f32_kernel<<<grdTok64, gemmBlk, 0, stream>>>(X, wqi, nullptr, Qb, T, 64, 64, 1.0f, 0);
    gemm_wmma_f32_kernel<<<grdTok64, gemmBlk, 0, stream>>>(X, wki, nullptr, Kb, T, 64, 64, 1.0f, 0);
    gemm_wmma_f32_kernel<<<grdTok64, gemmBlk, 0, stream>>>(X, wvi, nullptr, Vb, T, 64, 64, 1.0f, 0);

    attn_kernel<<<dim3(Bb * NHEAD), dim3(64), 0, stream>>>(Qb, Kb, Vb, seq, CTX);

    gemm_wmma_f32_kernel<<<grdTok64, gemmBlk, 0, stream>>>(CTX, woi, nullptr, Qb, T, 64, 64, 1.0f, 0);
    add_ln_kernel<<<gsz(T, 128), dim3(128), 0, stream>>>(X, Qb, ln1g + i * 64, ln1b + i * 64, X, T);

    gemm_wmma_f32_kernel<<<grdTok256, gemmBlk, 0, stream>>>(
        X, w1 + (long)i * 64 * 256, b1 + i * 256, HFF, T, 64, 256, 1.0f, 1);
    gemm_wmma_f32_kernel<<<grdTok64, gemmBlk, 0, stream>>>(
        HFF, w2 + (long)i * 256 * 64, b2 + i * 64, Qb, T, 256, 64, 1.0f, 0);
    add_ln_kernel<<<gsz(T, 128), dim3(128), 0, stream>>>(X, Qb, ln2g + i * 64, ln2b + i * 64, X, T);
  }

  // ---- final LN + gather ----
  final_kernel<<<gsz(Bb, 128), dim3(128), 0, stream>>>(X, lengths, lnfg, lnfb, (float*)d_out, Bb);
}